// BilinearInteraction_86766929313814
// MI455X (gfx1250) — compile-verified
//
#include <hip/hip_runtime.h>

typedef __attribute__((ext_vector_type(2))) float v2f;
typedef __attribute__((ext_vector_type(4))) float v4f;
typedef __attribute__((ext_vector_type(8))) float v8f;

#define NF   32          // fields
#define ND   64          // embed dim
#define NP   496         // NF*(NF-1)/2 pairs
#define LDX  68          // padded row stride for x / vid in LDS (multiple of 4 -> float4 aligned)
#define LDW  65          // padded row stride for W in LDS (odd -> conflict-free column reads)

__launch_bounds__(256)
__global__ void bilinear_interaction_kernel(const float* __restrict__ x,
                                            const float* __restrict__ W,
                                            float* __restrict__ out) {
    __shared__ float sW[ND * LDW];        // 64 x 64 (padded)   ~16.3 KB
    __shared__ float sX[NF * LDX];        // 32 x 64 (padded)   ~8.7 KB
    __shared__ float sV[NF * LDX];        // vid = x @ W        ~8.7 KB
    __shared__ unsigned char sPi[NP];
    __shared__ unsigned char sPj[NP];

    const int tid = threadIdx.x;
    const int b   = blockIdx.x;

    // ---- pair table: triu_indices(32, k=1) order (i-major, j ascending) ----
    for (int p = tid; p < NP; p += 256) {
        int i = 0, rem = p;
        while (rem >= (NF - 1 - i)) { rem -= (NF - 1 - i); ++i; }
        sPi[p] = (unsigned char)i;
        sPj[p] = (unsigned char)(i + 1 + rem);
    }

    // ---- stage W into LDS (4096 floats = 1024 float4, 4 iters of 256) ----
    {
        const v4f* Wv = (const v4f*)W;
#pragma unroll
        for (int it = 0; it < 4; ++it) {
            int idx4 = tid + it * 256;
            v4f v = Wv[idx4];
            int r = idx4 >> 4;            // 16 float4 per 64-wide row
            int c = (idx4 & 15) * 4;
            float* dst = &sW[r * LDW + c];
            dst[0] = v.x; dst[1] = v.y; dst[2] = v.z; dst[3] = v.w;
        }
    }
    // ---- stage x[b] into LDS (2048 floats = 512 float4, 2 iters) ----
    {
        const v4f* Xv = (const v4f*)(x + (size_t)b * (NF * ND));
#pragma unroll
        for (int it = 0; it < 2; ++it) {
            int idx4 = tid + it * 256;
            v4f v = Xv[idx4];
            int r = idx4 >> 4;
            int c = (idx4 & 15) * 4;      // LDX % 4 == 0 -> aligned float4 LDS store
            *(v4f*)&sX[r * LDX + c] = v;
        }
    }
    __syncthreads();

    // ---- vid = x[b] (32x64) @ W (64x64), exact fp32 via V_WMMA_F32_16X16X4_F32 ----
    // 8 tiles of 16x16 (2 row-tiles x 4 col-tiles), one tile per wave; K loop of 16 steps.
    {
        const int lane = tid & 31;
        const int wave = tid >> 5;        // 0..7
        const int lm   = lane & 15;
        const int half = lane >> 4;       // 0 or 1
        const int m0   = (wave & 1) * 16;
        const int n0   = (wave >> 1) * 16;

        v8f acc = {};
#pragma unroll
        for (int k0 = 0; k0 < ND; k0 += 4) {
            // A 16x4 layout: lanes 0-15 {K=k0,k0+1}, lanes 16-31 {K=k0+2,k0+3}
            const int ka = k0 + 2 * half;
            v2f a, bb;
            a.x  = sX[(m0 + lm) * LDX + ka];
            a.y  = sX[(m0 + lm) * LDX + ka + 1];
            // B 4x16 layout mirrors A: row K striped across lanes within a VGPR
            bb.x = sW[ka * LDW + n0 + lm];
            bb.y = sW[(ka + 1) * LDW + n0 + lm];
            acc = __builtin_amdgcn_wmma_f32_16x16x4_f32(
                    /*neg_a=*/false, a, /*neg_b=*/false, bb,
                    /*c_mod=*/(short)0, acc, /*reuse_a=*/false, /*reuse_b=*/false);
        }
        // C/D 16x16 f32 layout: VGPR v -> M = v + 8*half, N = lm
#pragma unroll
        for (int v = 0; v < 8; ++v) {
            sV[(m0 + v + 8 * half) * LDX + n0 + lm] = acc[v];
        }
    }
    __syncthreads();

    // ---- stream out[b, p, :] = x[b, i, :] * vid[b, j, :]  (496 x 64 floats) ----
    // 16 pairs per iteration (256 threads * float4 = 1024 contiguous floats), 31 iters.
    {
        float* ob = out + (size_t)b * (NP * ND);
        const int dg  = (tid & 15) * 4;   // dim offset within the 64-wide row
        const int prL = tid >> 4;         // local pair 0..15
#pragma unroll 4
        for (int it = 0; it < 31; ++it) {
            int p = it * 16 + prL;
            int i = sPi[p];
            int j = sPj[p];
            v4f xv = *(const v4f*)&sX[i * LDX + dg];
            v4f vv = *(const v4f*)&sV[j * LDX + dg];
            v4f o  = xv * vv;
            __builtin_nontemporal_store(o, (v4f*)(ob + p * ND + dg));
        }
    }
}

extern "C" void kernel_launch(void* const* d_in, const int* in_sizes, int n_in,
                              void* d_out, int out_size, void* d_ws, size_t ws_size,
                              hipStream_t stream) {
    const float* x = (const float*)d_in[0];   // [B, 32, 64] f32
    const float* W = (const float*)d_in[1];   // [64, 64] f32
    float* out = (float*)d_out;               // [B, 496, 64] f32

    const int batch = in_sizes[0] / (NF * ND);  // 4096
    bilinear_interaction_kernel<<<batch, 256, 0, stream>>>(x, W, out);
}